// C4WithSyscalls_62380105007287
// MI455X (gfx1250) — compile-verified
//
#include <hip/hip_runtime.h>
#include <hip/hip_bf16.h>
#include <math.h>

// ---- static configuration mirrored from the reference ----
#define C4_MEM   16384
#define C4_OUT   64
#define C4_NCTX  (C4_MEM + 5 + C4_OUT)   // 16453

// ---------------------------------------------------------------------
// Soft-gate helpers (exact reference formulas, GATE_SCALE = 20)
// ---------------------------------------------------------------------
__device__ __forceinline__ float c4_sigmoid(float x) { return 1.0f / (1.0f + expf(-x)); }
__device__ __forceinline__ float c4_silu(float x)    { return x / (1.0f + expf(-x)); }
__device__ __forceinline__ float c4_sthr(float x) {               // silu_threshold
    float d = 20.0f * x;
    return (c4_silu(d + 10.0f) - c4_silu(d - 10.0f)) / 20.0f;
}
__device__ __forceinline__ float c4_eq(float a, float b) {        // eq_gate
    float d = a - b;
    return c4_sthr(d + 0.5f) * c4_sthr(-d + 0.5f);
}

// =====================================================================
// Fused per-row kernel. Block b owns batch row b:
//   1) async DMA global row -> LDS (CDNA5 ASYNCcnt path, no VGPR data)
//   2) wave 0 runs the 16 VM steps entirely in LDS; the 15-expert
//      gating is spread across lanes 0..14 and reduced with shfl_xor,
//      so every lane keeps identical PC/SP/BP/AX state
//   3) async DMA LDS row -> global d_out
// The softmax attention collapses to an exact one-hot in f32 (score
// margin 200/sqrt(20) => runner-up weight ~4e-20 << f32 ulp), so
// attend == gather at int(clip(addr)) and the soft scatter is a
// single-cell lerp.
// =====================================================================
__global__ __launch_bounds__(256) void c4_fused(const float* __restrict__ src,
                                                float* __restrict__ dst,
                                                const float* __restrict__ outbuf,
                                                const float* __restrict__ ax0,
                                                const int* __restrict__ pc0,
                                                const int* __restrict__ sp0,
                                                const int* __restrict__ bp0,
                                                const int* __restrict__ olen,
                                                const int* __restrict__ nsp) {
    __shared__ __align__(16) float row[C4_MEM];   // 64 KB row staging
    const int b = blockIdx.x;
    const unsigned tid = threadIdx.x;
    const unsigned long long sbase =
        (unsigned long long)(const void*)(src + (size_t)b * C4_MEM);
    const unsigned long long dbase =
        (unsigned long long)(void*)(dst + (size_t)b * C4_MEM);

    // ---- 1) async copy: global row -> LDS (16 rounds x 256 lanes x 16 B) ----
#pragma unroll
    for (int r = 0; r < 16; ++r) {
        unsigned off = (r * 256u + tid) * 16u;
        // low 32 bits of the flat shared address == LDS byte address
        unsigned lds = (unsigned)(unsigned long long)(const void*)
                       ((const char*)row + off);
        unsigned long long ga = sbase + off;
        asm volatile("global_load_async_to_lds_b128 %0, %1, off"
                     :: "v"(lds), "v"(ga) : "memory");
    }
    asm volatile("s_wait_asynccnt 0" ::: "memory");
    __syncthreads();                                // LDS row fully populated

    // ---- 2) interpreter on wave 0, experts spread across lanes ----
    if (tid < 32u) {
        const int i = (int)tid;                     // lane id == expert id
        const int nsteps = nsp[0];
        float pc = (float)pc0[b], sp = (float)sp0[b], bp = (float)bp0[b];
        float ax = ax0[b];
        const float olf = (float)olen[b];
        const float* op = outbuf + (size_t)b * C4_OUT;

        // per-lane opcode value (IMM,LEA,ADD,SUB,MUL,DIV,MOD,SHL,SHR,EQ,NE,LT,GT,LE,GE)
        const float opval =
            (i==0)?1.f:(i==1)?2.f:(i==2)?9.f:(i==3)?10.f:(i==4)?11.f:(i==5)?12.f:
            (i==6)?13.f:(i==7)?14.f:(i==8)?15.f:(i==9)?3.f:(i==10)?4.f:(i==11)?5.f:
            (i==12)?6.f:(i==13)?7.f:(i==14)?8.f:0.f;
        const bool active = (i < 15);
        const bool pops   = (i >= 2) && (i < 15);   // POP_MASK

        for (int s = 0; s < nsteps; ++s) {
            // attend(context, pc) == hard gather (uniform LDS broadcast)
            float cp = fminf(fmaxf(pc, 0.0f), (float)(C4_NCTX - 1));
            int aip = (int)cp;
            float inst;
            if (aip < C4_MEM) inst = row[aip];
            else { int r = aip - C4_MEM;
                   inst = (r==0)?pc:(r==1)?sp:(r==2)?bp:(r==3)?ax:(r==4)?olf:op[r-5]; }
            float imm = floorf(inst / 256.0f);
            float opc = inst - imm * 256.0f;

            // attend(context, sp) == hard gather
            float cs = fminf(fmaxf(sp, 0.0f), (float)(C4_NCTX - 1));
            int ais = (int)cs;
            float a;
            if (ais < C4_MEM) a = row[ais];
            else { int r = ais - C4_MEM;
                   a = (r==0)?pc:(r==1)?sp:(r==2)?bp:(r==3)?ax:(r==4)?olf:op[r-5]; }

            // shared expert scalars (computed SIMD on all lanes)
            float bb = ax;
            float safeb = (fabsf(bb) < 1e-6f) ? 1e-6f : bb;
            float dv = a / safeb;
            float md = a - safeb * floorf(dv);
            float sh = fminf(fmaxf(bb, 0.0f), 31.0f);
            float eqg = c4_eq(a, bb);
            float lt = c4_sigmoid(20.0f * (bb - a - 0.5f));
            float gt = c4_sigmoid(20.0f * (a - bb - 0.5f));

            // lane i's expert output (cndmask chain, no divergence)
            float out =
                (i==0)?imm:(i==1)?bp+imm:(i==2)?a+bb:(i==3)?a-bb:(i==4)?a*bb:
                (i==5)?dv:(i==6)?md:(i==7)?a*exp2f(sh):(i==8)?a*exp2f(-sh):
                (i==9)?eqg:(i==10)?1.0f-eqg:(i==11)?lt:(i==12)?gt:
                (i==13)?1.0f-gt:1.0f-lt;

            // lane i's gate; lanes >= 15 contribute zero
            float g    = active ? c4_eq(opc, opval) : 0.0f;
            float gout = g * out;
            float gpop = pops ? g : 0.0f;

            // butterfly reduction -> identical sums on every lane
            float gs = g, ac = gout, pp = gpop;
#pragma unroll
            for (int mdel = 16; mdel >= 1; mdel >>= 1) {
                gs += __shfl_xor(gs, mdel);
                ac += __shfl_xor(ac, mdel);
                pp += __shfl_xor(pp, mdel);
            }

            float nax = ac + (1.0f - gs) * ax;

            // soft scatter == single-cell lerp at old stack-top address
            // (uniform address, uniform value: whole wave writes the same)
            if (ais < C4_MEM) row[ais] = row[ais] + pp * (nax - row[ais]);

            ax = nax;
            sp = sp + 8.0f * pp;
            pc = pc + 8.0f;
        }
    }
    __syncthreads();                                // row finalized in LDS

    // ---- 3) async copy: LDS row -> global d_out ----
#pragma unroll
    for (int r = 0; r < 16; ++r) {
        unsigned off = (r * 256u + tid) * 16u;
        unsigned lds = (unsigned)(unsigned long long)(const void*)
                       ((const char*)row + off);
        unsigned long long ga = dbase + off;
        asm volatile("global_store_async_from_lds_b128 %0, %1, off"
                     :: "v"(ga), "v"(lds) : "memory");
    }
    asm volatile("s_wait_asynccnt 0" ::: "memory");
}

extern "C" void kernel_launch(void* const* d_in, const int* in_sizes, int n_in,
                              void* d_out, int out_size, void* d_ws, size_t ws_size,
                              hipStream_t stream) {
    const float* memory  = (const float*)d_in[0];   // [B, MEM]
    const float* output  = (const float*)d_in[1];   // [B, OUT]
    const float* ax      = (const float*)d_in[2];   // [B]
    const int*   pc      = (const int*)d_in[3];
    const int*   sp      = (const int*)d_in[4];
    const int*   bp      = (const int*)d_in[5];
    const int*   out_len = (const int*)d_in[6];
    const int*   nsteps  = (const int*)d_in[7];     // scalar
    float* out = (float*)d_out;

    const int B = in_sizes[2];                      // 1024 rows
    c4_fused<<<B, 256, 0, stream>>>(memory, out, output, ax, pc, sp, bp,
                                    out_len, nsteps);
    (void)d_ws; (void)ws_size; (void)n_in; (void)out_size;
}